// EmbedingLayer_39994735460380
// MI455X (gfx1250) — compile-verified
//
#include <hip/hip_runtime.h>
#include <math.h>

typedef __attribute__((ext_vector_type(2))) float v2f;
typedef __attribute__((ext_vector_type(4))) float v4f;
typedef __attribute__((ext_vector_type(8))) float v8f;

#define B_  512
#define S_  256
#define H_  768
#define P_  32
#define K1_ (3 * H_)   // 2304
#define K2_ (2 * H_)   // 1536

// ---------------------------------------------------------------------------
// Kernel 1: masked mean-pool over P positions + pack concatenated GEMM inputs.
// One 192-thread block (6 waves) per batch element; each thread owns a
// contiguous float4 slice of H (192*4 = 768) so every gather is a coalesced
// global_load_b128. This phase carries ~100 MB of traffic and sets the
// kernel's bandwidth floor, so wide loads matter most here.
//   Xc[b, 0:768]       = mean_{valid p} proc[b, p, :]
//   Xc[b, 768:1536]    = proc[b, 0, :]
//   Xc[b, 1536:2304]   = phys[b, 0, :]
//   Xm[b, 0:768]       = mean_{valid p} micro[b, p, :]
//   Xm[b, 768:1536]    = micro[b, 0, :]
// ---------------------------------------------------------------------------
__global__ void __launch_bounds__(192)
pool_pack_kernel(const float* __restrict__ phys,
                 const float* __restrict__ proc,
                 const float* __restrict__ micro,
                 const int* __restrict__ pos,
                 float* __restrict__ Xc,
                 float* __restrict__ Xm) {
  const int b = blockIdx.x;
  const int t = threadIdx.x;  // 0..191
  const int c = t * 4;        // float4 column slice

  __shared__ int s_pos[P_];
  __shared__ float s_inv;
  if (t < P_) s_pos[t] = pos[b * P_ + t];
  __syncthreads();
  if (t == 0) {
    int cnt = 0;
    for (int i = 0; i < P_; ++i) cnt += (s_pos[i] != -1) ? 1 : 0;
    s_inv = 1.0f / (float)cnt;  // cnt >= 1 (position 0 never dropped)
  }
  __syncthreads();
  const float inv = s_inv;

  const float* procB  = proc  + (size_t)b * S_ * H_;
  const float* microB = micro + (size_t)b * S_ * H_;
  const float* physB  = phys  + (size_t)b * S_ * H_;
  float* xc = Xc + (size_t)b * K1_;
  float* xm = Xm + (size_t)b * K2_;

  v4f sp = {}, sm = {};
  for (int i = 0; i < P_; ++i) {
    const int p = s_pos[i];
    if (p != -1) {
      sp += *(const v4f*)(procB  + (size_t)p * H_ + c);
      sm += *(const v4f*)(microB + (size_t)p * H_ + c);
    }
  }
  *(v4f*)(xc + c)          = sp * inv;                      // pooled_proc
  *(v4f*)(xc + H_ + c)     = *(const v4f*)(procB + c);      // proc_sent (row 0)
  *(v4f*)(xc + 2*H_ + c)   = *(const v4f*)(physB + c);      // phys_sent (row 0)
  *(v4f*)(xm + c)          = sm * inv;                      // pooled_micro
  *(v4f*)(xm + H_ + c)     = *(const v4f*)(microB + c);     // micro_sent (row 0)
}

// ---------------------------------------------------------------------------
// Kernel 2: Y(M x 768) = X(M x K) * W^T + bias, via V_WMMA_F32_16X16X4_F32.
// W is stored (N x K) row-major, so WMMA B element B[k,n] = W[n*K + k] -- a
// contiguous K-pair per lane (one b64 load).
//
// Per-wave tile: 32 rows (M) x 64 cols (N) => 8 x v8f accumulators.
// Per K-step: 2 A loads + 4 B loads feed 8 WMMAs (0.75 loads/WMMA), so both
// A and B operands are reused across the tile. blockDim = (32, 4).
//
// Lane layouts (ISA 7.12.2, 32-bit, wave32):
//   A 16x4 : lane l holds M = l%16, K = k0 + 2*(l/16) + {0,1}  -> one float2
//   B 4x16 : lane l holds N = l%16, K = k0 + 2*(l/16) + {0,1}  -> one float2
//   C 16x16: VGPR r, lane l ->  M = r + 8*(l/16), N = l%16
// ---------------------------------------------------------------------------
__global__ void __launch_bounds__(128)
gemm_wmma_f32(const float* __restrict__ X,     // M x K
              const float* __restrict__ W,     // N x K  (row-major)
              const float* __restrict__ bias,  // N
              float* __restrict__ Y,           // M x N  (N == 768)
              int K) {
  const int N    = H_;
  const int lane = threadIdx.x;        // 0..31 (wave32)
  const int half = lane >> 4;          // 0 or 1
  const int lm   = lane & 15;

  const int m0 = blockIdx.x * 32;
  const int n0 = (blockIdx.y * 4 + threadIdx.y) * 64;

  v8f accA0 = {}, accA1 = {}, accA2 = {}, accA3 = {};
  v8f accB0 = {}, accB1 = {}, accB2 = {}, accB3 = {};

  const float* a0p = X + (size_t)(m0 +  0 + lm) * K + 2 * half;
  const float* a1p = X + (size_t)(m0 + 16 + lm) * K + 2 * half;
  const float* w0p = W + (size_t)(n0 +  0 + lm) * K + 2 * half;
  const float* w1p = W + (size_t)(n0 + 16 + lm) * K + 2 * half;
  const float* w2p = W + (size_t)(n0 + 32 + lm) * K + 2 * half;
  const float* w3p = W + (size_t)(n0 + 48 + lm) * K + 2 * half;

#pragma unroll 2
  for (int k = 0; k < K; k += 4) {
    const v2f a0 = *(const v2f*)(a0p + k);
    const v2f a1 = *(const v2f*)(a1p + k);
    const v2f w0 = *(const v2f*)(w0p + k);
    const v2f w1 = *(const v2f*)(w1p + k);
    const v2f w2 = *(const v2f*)(w2p + k);
    const v2f w3 = *(const v2f*)(w3p + k);
    accA0 = __builtin_amdgcn_wmma_f32_16x16x4_f32(false, a0, false, w0, (short)0, accA0, false, false);
    accA1 = __builtin_amdgcn_wmma_f32_16x16x4_f32(false, a0, false, w1, (short)0, accA1, false, false);
    accA2 = __builtin_amdgcn_wmma_f32_16x16x4_f32(false, a0, false, w2, (short)0, accA2, false, false);
    accA3 = __builtin_amdgcn_wmma_f32_16x16x4_f32(false, a0, false, w3, (short)0, accA3, false, false);
    accB0 = __builtin_amdgcn_wmma_f32_16x16x4_f32(false, a1, false, w0, (short)0, accB0, false, false);
    accB1 = __builtin_amdgcn_wmma_f32_16x16x4_f32(false, a1, false, w1, (short)0, accB1, false, false);
    accB2 = __builtin_amdgcn_wmma_f32_16x16x4_f32(false, a1, false, w2, (short)0, accB2, false, false);
    accB3 = __builtin_amdgcn_wmma_f32_16x16x4_f32(false, a1, false, w3, (short)0, accB3, false, false);
  }

  // Epilogue: fused bias add, unnormalized result to Y.
  const int col = n0 + lm;
  const float bv0 = bias[col +  0];
  const float bv1 = bias[col + 16];
  const float bv2 = bias[col + 32];
  const float bv3 = bias[col + 48];
#pragma unroll
  for (int r = 0; r < 8; ++r) {
    const int rowA = m0 + r + 8 * half;
    float* yA = Y + (size_t)rowA * N;
    yA[col +  0] = accA0[r] + bv0;
    yA[col + 16] = accA1[r] + bv1;
    yA[col + 32] = accA2[r] + bv2;
    yA[col + 48] = accA3[r] + bv3;
    float* yB = yA + (size_t)16 * N;
    yB[col +  0] = accB0[r] + bv0;
    yB[col + 16] = accB1[r] + bv1;
    yB[col + 32] = accB2[r] + bv2;
    yB[col + 48] = accB3[r] + bv3;
  }
}

// ---------------------------------------------------------------------------
// Kernel 3: in-place row-wise L2 normalize (rows of length H_=768).
// grid = total rows (2*B), block = 256 threads (3 elements each).
// ---------------------------------------------------------------------------
__global__ void l2norm_kernel(float* __restrict__ Y) {
  const int row = blockIdx.x;
  const int t   = threadIdx.x;  // 0..255
  float* y = Y + (size_t)row * H_;

  const float v0 = y[t];
  const float v1 = y[t + 256];
  const float v2 = y[t + 512];

  __shared__ float red[256];
  red[t] = v0 * v0 + v1 * v1 + v2 * v2;
  __syncthreads();
  for (int off = 128; off > 0; off >>= 1) {
    if (t < off) red[t] += red[t + off];
    __syncthreads();
  }

  __shared__ float s_scale;
  if (t == 0) {
    const float n = sqrtf(red[0]);
    s_scale = 1.0f / fmaxf(n, 1e-12f);
  }
  __syncthreads();
  const float sc = s_scale;

  y[t]       = v0 * sc;
  y[t + 256] = v1 * sc;
  y[t + 512] = v2 * sc;
}

// ---------------------------------------------------------------------------
extern "C" void kernel_launch(void* const* d_in, const int* in_sizes, int n_in,
                              void* d_out, int out_size, void* d_ws, size_t ws_size,
                              hipStream_t stream) {
  const float* phys  = (const float*)d_in[0];  // (B,S,H)
  const float* proc  = (const float*)d_in[1];  // (B,S,H)
  const float* micro = (const float*)d_in[2];  // (B,S,H)
  const int*   pos   = (const int*)d_in[3];    // (B,P) int32
  const float* W1    = (const float*)d_in[4];  // (H, 3H)
  const float* b1    = (const float*)d_in[5];  // (H,)
  const float* W2    = (const float*)d_in[6];  // (H, 2H)
  const float* b2    = (const float*)d_in[7];  // (H,)

  float* out = (float*)d_out;                  // [B*H | B*H]

  // Workspace: Xc (B x 3H) then Xm (B x 2H)  => ~7.9 MB total.
  float* Xc = (float*)d_ws;
  float* Xm = Xc + (size_t)B_ * K1_;

  float* Yc = out;                     // normalized_combined (in-place later)
  float* Ym = out + (size_t)B_ * H_;   // normalized_micro

  // 1) pool + pack (b128 gathers)
  pool_pack_kernel<<<B_, 192, 0, stream>>>(phys, proc, micro, pos, Xc, Xm);

  // 2) two f32 WMMA GEMMs (bias fused), 32x64 tile per wave
  dim3 block(32, 4);                       // 4 wave32 per block
  dim3 grid(B_ / 32, H_ / (64 * 4));       // (16, 3)
  gemm_wmma_f32<<<grid, block, 0, stream>>>(Xc, W1, b1, Yc, K1_);
  gemm_wmma_f32<<<grid, block, 0, stream>>>(Xm, W2, b2, Ym, K2_);

  // 3) in-place L2 normalize all 2*B rows
  l2norm_kernel<<<2 * B_, 256, 0, stream>>>(out);
}